// PolicyNet_17248588660955
// MI455X (gfx1250) — compile-verified
//
#include <hip/hip_runtime.h>
#include <math.h>

// CDNA5 wave32 WMMA operand types for V_WMMA_F32_16X16X4_F32:
// A = 16x4 f32 (64 vals / 32 lanes = 2/lane), B = 4x16 f32, C/D = 16x16 f32 (8/lane)
typedef float v2f __attribute__((ext_vector_type(2)));
typedef float v8f __attribute__((ext_vector_type(8)));

#define L_LM    1000
#define IN_DIM  4003
#define TILES   63            // ceil(1000/16); 63*16 = 1008 (last 8 rows padded)
#define NWAVES  8
#define BLOCK   (NWAVES * 32)
#define E1_STR  65            // padded row stride (floats) to avoid LDS bank conflicts

__launch_bounds__(BLOCK)
__global__ void policy_rows01_kernel(
    const float* __restrict__ obs,
    const float* __restrict__ W1,  const float* __restrict__ b1,
    const float* __restrict__ W2,  const float* __restrict__ b2,
    const float* __restrict__ Wl1, const float* __restrict__ bl1,
    const float* __restrict__ Wl2, const float* __restrict__ bl2,
    const float* __restrict__ Wi,  const float* __restrict__ bi,
    const float* __restrict__ Wa,  const float* __restrict__ ba,
    float* __restrict__ out,       // [4]
    float* __restrict__ e_ws)      // [2][1008][32] f32 in d_ws
{
    __shared__ float a1_lds[32];                 // a1, later reused for lam
    __shared__ float a_lds[32];                  // agent embedding
    __shared__ float e1_lds[NWAVES][16 * E1_STR]; // per-wave 16x64 fc1 tile (padded)
    __shared__ float score_lds[TILES * 16];      // 1008 scores
    __shared__ float red_lds[NWAVES];
    __shared__ float part_lds[NWAVES][32];
    __shared__ float h_lds[64];
    __shared__ float stat_lds[2];                // [0]=max, [1]=sumexp

    const int b      = blockIdx.x;               // batch row 0 or 1
    const int tid    = threadIdx.x;
    const int lane   = tid & 31;
    const int wv     = tid >> 5;
    const int lanelo = lane & 15;
    const bool hihalf = (lane >= 16);
    const int krA    = hihalf ? 2 : 0;           // K rows held by this half-wave

    const float* o = obs + (size_t)b * IN_DIM;
    float* e_row = e_ws + (size_t)b * (TILES * 16) * 32;

    // ---------------- Phase A: agent embedding (3 -> 32 -> 32) ----------------
    if (tid < 32) {
        float s = b1[tid];
        #pragma unroll
        for (int k = 0; k < 3; ++k) s += o[k] * W1[k * 32 + tid];
        a1_lds[tid] = fmaxf(s, 0.f);
    }
    __syncthreads();
    if (tid < 32) {
        float s = b2[tid];
        #pragma unroll
        for (int k = 0; k < 32; ++k) s += a1_lds[k] * W2[k * 32 + tid];
        a_lds[tid] = fmaxf(s, 0.f);
    }
    __syncthreads();

    // ---------------- Preload B operands (weights) per lane ----------------
    // Wl1 [4,64]: B tile nt covers cols nt*16+lanelo; VGPR0=K{0|2}, VGPR1=K{1|3}
    v2f B1[4];
    float cb1[4];
    #pragma unroll
    for (int nt = 0; nt < 4; ++nt) {
        const int n = nt * 16 + lanelo;
        B1[nt].x = Wl1[(krA    ) * 64 + n];
        B1[nt].y = Wl1[(krA + 1) * 64 + n];
        cb1[nt]  = bl1[n];
    }
    // Wl2 [64,32]: 16 K-steps x 2 N-tiles
    v2f B2[16][2];
    #pragma unroll
    for (int k = 0; k < 16; ++k) {
        #pragma unroll
        for (int nt = 0; nt < 2; ++nt) {
            const int n = nt * 16 + lanelo;
            B2[k][nt].x = Wl2[(4 * k + krA    ) * 32 + n];
            B2[k][nt].y = Wl2[(4 * k + krA + 1) * 32 + n];
        }
    }
    float cb2[2];
    cb2[0] = bl2[lanelo];
    cb2[1] = bl2[16 + lanelo];

    // ---------------- Phase B: landmark MLP tiles via WMMA ----------------
    float* my_e1 = &e1_lds[wv][0];
    for (int t = wv; t < TILES; t += NWAVES) {
        const int l = t * 16 + lanelo;           // this lane's landmark row (M = lanelo)
        const bool valid = (l < L_LM);

        // A operand: lm[l] = (pos.x, pos.y, info.x, info.y); K=0,1 -> pos; K=2,3 -> info
        v2f A;
        if (valid) {
            const float* src = hihalf ? (o + 3 + 2 * l)              // info (K=2,3)
                                      : (o + 3 + 2 * L_LM + 2 * l);  // pos  (K=0,1)
            A.x = src[0];
            A.y = src[1];
        } else {
            A.x = 0.f; A.y = 0.f;
        }

        // fc1: e1 = relu(A[16x4] @ Wl1[4x64] + bl1) -- 4 WMMAs (one per 16-col tile)
        #pragma unroll
        for (int nt = 0; nt < 4; ++nt) {
            v8f c;
            #pragma unroll
            for (int r = 0; r < 8; ++r) c[r] = cb1[nt];
            v8f d = __builtin_amdgcn_wmma_f32_16x16x4_f32(
                        false, A, false, B1[nt], (short)0, c, false, false);
            #pragma unroll
            for (int r = 0; r < 8; ++r) {
                const int m = r + (hihalf ? 8 : 0);
                my_e1[m * E1_STR + nt * 16 + lanelo] = fmaxf(d[r], 0.f);
            }
        }
        // in-wave LDS RAW ordering before re-reading the tile transposed
        asm volatile("s_wait_dscnt 0" ::: "memory");

        // fc2: e = relu(e1[16x64] @ Wl2[64x32] + bl2) -- 16 chained K=4 steps x 2 N-tiles
        v8f acc0, acc1;
        #pragma unroll
        for (int r = 0; r < 8; ++r) { acc0[r] = cb2[0]; acc1[r] = cb2[1]; }
        #pragma unroll
        for (int k = 0; k < 16; ++k) {
            v2f A2;
            A2.x = my_e1[lanelo * E1_STR + 4 * k + krA];
            A2.y = my_e1[lanelo * E1_STR + 4 * k + krA + 1];
            acc0 = __builtin_amdgcn_wmma_f32_16x16x4_f32(
                       false, A2, false, B2[k][0], (short)0, acc0, false, false);
            acc1 = __builtin_amdgcn_wmma_f32_16x16x4_f32(
                       false, A2, false, B2[k][1], (short)0, acc1, false, false);
        }
        // relu + store e tile (D layout -> row-major [l][32]) into L2-resident workspace
        #pragma unroll
        for (int r = 0; r < 8; ++r) {
            const int m  = r + (hihalf ? 8 : 0);
            const int lr = t * 16 + m;
            e_row[lr * 32 +      lanelo] = fmaxf(acc0[r], 0.f);
            e_row[lr * 32 + 16 + lanelo] = fmaxf(acc1[r], 0.f);
        }
    }
    __threadfence_block();
    __syncthreads();

    // ---------------- Phase C: attention scores ----------------
    for (int l = tid; l < L_LM; l += BLOCK) {
        const float* ev = e_row + l * 32;
        float s = 0.f;
        #pragma unroll
        for (int d = 0; d < 32; ++d) s += a_lds[d] * ev[d];
        score_lds[l] = s * 0.25f;
    }
    __syncthreads();

    // ---------------- Phase D: softmax stats over 1000 landmarks ----------------
    float m = -1e30f;
    for (int l = tid; l < L_LM; l += BLOCK) m = fmaxf(m, score_lds[l]);
    #pragma unroll
    for (int off = 16; off > 0; off >>= 1) m = fmaxf(m, __shfl_xor(m, off, 32));
    if (lane == 0) red_lds[wv] = m;
    __syncthreads();
    if (tid == 0) {
        float mm = red_lds[0];
        for (int w = 1; w < NWAVES; ++w) mm = fmaxf(mm, red_lds[w]);
        stat_lds[0] = mm;
    }
    __syncthreads();
    const float gmax = stat_lds[0];

    float se = 0.f;
    for (int l = tid; l < L_LM; l += BLOCK) se += expf(score_lds[l] - gmax);
    #pragma unroll
    for (int off = 16; off > 0; off >>= 1) se += __shfl_xor(se, off, 32);
    if (lane == 0) red_lds[wv] = se;
    __syncthreads();
    if (tid == 0) {
        float ss = 0.f;
        for (int w = 0; w < NWAVES; ++w) ss += red_lds[w];
        stat_lds[1] = ss;
    }
    __syncthreads();
    const float invsum = 1.f / stat_lds[1];

    // ---------------- lam = relu(att @ e) ----------------
    {
        const int d = tid & 31;
        float p = 0.f;
        for (int l = wv; l < L_LM; l += NWAVES)
            p += expf(score_lds[l] - gmax) * e_row[l * 32 + d];
        part_lds[wv][d] = p;
    }
    __syncthreads();
    if (tid < 32) {
        float p = 0.f;
        #pragma unroll
        for (int g = 0; g < NWAVES; ++g) p += part_lds[g][tid];
        a1_lds[tid] = fmaxf(p * invsum, 0.f);    // lam (reuse a1_lds)
    }
    __syncthreads();

    // ---------------- Fusion head + action ----------------
    if (tid < 64) {
        float s = bi[tid];
        #pragma unroll
        for (int k = 0; k < 32; ++k) s += a_lds[k]  * Wi[k * 64 + tid];
        #pragma unroll
        for (int k = 0; k < 32; ++k) s += a1_lds[k] * Wi[(32 + k) * 64 + tid];
        h_lds[tid] = fmaxf(s, 0.f);
    }
    __syncthreads();
    if (tid < 2) {
        float s = ba[tid];
        #pragma unroll
        for (int k = 0; k < 64; ++k) s += h_lds[k] * Wa[k * 2 + tid];
        const float act   = tanhf(s);
        const float scale = (b == 0) ? 3.0f : 0.1f;
        out[b * 2 + tid] = act * scale;
    }
}

extern "C" void kernel_launch(void* const* d_in, const int* in_sizes, int n_in,
                              void* d_out, int out_size, void* d_ws, size_t ws_size,
                              hipStream_t stream) {
    const float* obs = (const float*)d_in[0];
    const float* W1  = (const float*)d_in[1];
    const float* b1  = (const float*)d_in[2];
    const float* W2  = (const float*)d_in[3];
    const float* b2  = (const float*)d_in[4];
    const float* Wl1 = (const float*)d_in[5];
    const float* bl1 = (const float*)d_in[6];
    const float* Wl2 = (const float*)d_in[7];
    const float* bl2 = (const float*)d_in[8];
    const float* Wi  = (const float*)d_in[9];
    const float* bi  = (const float*)d_in[10];
    const float* Wa  = (const float*)d_in[11];
    const float* ba  = (const float*)d_in[12];

    float* out  = (float*)d_out;
    float* e_ws = (float*)d_ws;   // needs 2 * 1008 * 32 * 4 = 258048 bytes

    // Output depends only on batch rows 0 and 1 -> one workgroup per row.
    policy_rows01_kernel<<<dim3(2), dim3(BLOCK), 0, stream>>>(
        obs, W1, b1, W2, b2, Wl1, bl1, Wl2, bl2, Wi, bi, Wa, ba, out, e_ws);
}